// UIGen_63058709840320
// MI455X (gfx1250) — compile-verified
//
#include <hip/hip_runtime.h>

#define LSEQ 2048
#define IVOC 32000
#define DMOD 256
#define NPART 125   // 32000 / 256 K-chunks for phase A

typedef float v2f __attribute__((ext_vector_type(2)));
typedef float v4f __attribute__((ext_vector_type(4)));
typedef float v8f __attribute__((ext_vector_type(8)));

// ---------------------------------------------------------------------------
// Phase A: partial GEMV — rows L-2 and L-1 of x times DE [32000 x 256].
// Each block handles a 256-wide K chunk; thread t owns output column t.
// DE reads are fully coalesced; x row values are wave-uniform broadcasts.
// No atomics: per-block partials -> deterministic.
// ---------------------------------------------------------------------------
__global__ __launch_bounds__(256) void phaseA(const float* __restrict__ x,
                                              const float* __restrict__ DE,
                                              float* __restrict__ partial) {
  const int t = threadIdx.x;
  const int b = blockIdx.x;
  const float* x0 = x + (size_t)(LSEQ - 2) * IVOC;
  const float* x1 = x + (size_t)(LSEQ - 1) * IVOC;
  const int i0 = b * 256;
  float acc0 = 0.0f, acc1 = 0.0f;
#pragma unroll 4
  for (int i = 0; i < 256; ++i) {
    const int k = i0 + i;
    const float w = DE[(size_t)k * DMOD + t];
    acc0 = fmaf(x0[k], w, acc0);
    acc1 = fmaf(x1[k], w, acc1);
  }
  partial[b * (2 * DMOD) + t]        = acc0;
  partial[b * (2 * DMOD) + DMOD + t] = acc1;
}

// ---------------------------------------------------------------------------
// Phase B: reduce partials -> posEmbed rows L-2/L-1 (+PE), then the whole
// dense1 / pool2 / attention / weighted chain for just those two rows.
// Emits s = weighted[L-2] + weighted[L-1]  (256 floats).
// ---------------------------------------------------------------------------
__global__ __launch_bounds__(256) void phaseB(const float* __restrict__ partial,
                                              const float* __restrict__ PE,
                                              const float* __restrict__ D1,
                                              const float* __restrict__ Att,
                                              float* __restrict__ sOut) {
  const int t = threadIdx.x;
  __shared__ float pe0[DMOD], pe1[DMOD], de0[DMOD], de1[DMOD];

  float a0 = PE[(size_t)(LSEQ - 2) * DMOD + t];
  float a1 = PE[(size_t)(LSEQ - 1) * DMOD + t];
  for (int b = 0; b < NPART; ++b) {
    a0 += partial[b * (2 * DMOD) + t];
    a1 += partial[b * (2 * DMOD) + DMOD + t];
  }
  pe0[t] = a0; pe1[t] = a1;
  __syncthreads();

  const float invLm1 = 1.0f / (float)(LSEQ - 1);  // PA column L-2 divisor
  const float invL   = 1.0f / (float)LSEQ;        // PA column L-1 divisor

  // dense1 rows L-2/L-1 : concat(posEmbed, pool1) @ D1
  float d0 = 0.0f, d1 = 0.0f;
  for (int c = 0; c < DMOD; ++c) {
    const float p0c = pe0[c], p1c = pe1[c];
    const float q0c = (p0c + p1c) * invLm1;  // pool1[L-2]
    const float q1c = p1c * invL;            // pool1[L-1]
    const float w1 = D1[c * DMOD + t];
    const float w2 = D1[(DMOD + c) * DMOD + t];
    d0 = fmaf(p0c, w1, fmaf(q0c, w2, d0));
    d1 = fmaf(p1c, w1, fmaf(q1c, w2, d1));
  }
  de0[t] = d0; de1[t] = d1;
  __syncthreads();

  // attention rows L-2/L-1 : concat(dense1, pool2) @ Att
  float t0 = 0.0f, t1 = 0.0f;
  for (int c = 0; c < DMOD; ++c) {
    const float e0c = de0[c], e1c = de1[c];
    const float r0c = (e0c + e1c) * invLm1;  // pool2[L-2]
    const float r1c = e1c * invL;            // pool2[L-1]
    const float w1 = Att[c * DMOD + t];
    const float w2 = Att[(DMOD + c) * DMOD + t];
    t0 = fmaf(e0c, w1, fmaf(r0c, w2, t0));
    t1 = fmaf(e1c, w1, fmaf(r1c, w2, t1));
  }
  sOut[t] = d0 * t0 + d1 * t1;  // weighted[L-2] + weighted[L-1]
}

// ---------------------------------------------------------------------------
// Phase C: out[0:32000] = s @ D2 via V_WMMA_F32_16X16X4_F32 (exact f32).
// A is a padded 16x256 matrix whose row 0 is s (rows 1..15 zero).
// Per ISA 16x4 f32 A layout: lane M = lane%16, VGPR0/1 hold K={0,1}
// (lanes 0-15) or K={2,3} (lanes 16-31) -> only lanes 0 and 16 carry data.
// B layout mirrors it: lane N = lane%16, VGPR0/1 = K {0,1}/{2,3}.
// C layout: VGPR0 lanes 0-15 == row M=0 -> the GEMV result.
//
// s is fetched with one uniform s_load_b128 per K-step; the A fragment is
// built with a hoisted multiplicative row mask so every load is demanded
// unconditionally -> no predicated loads, no saveexec, EXEC stays all-ones
// through the whole WMMA loop.
// ---------------------------------------------------------------------------
__global__ __launch_bounds__(256) void phaseC(const float* __restrict__ s,
                                              const float* __restrict__ D2,
                                              float* __restrict__ out) {
  const int wave  = threadIdx.x >> 5;          // 0..7
  const int lane  = threadIdx.x & 31;
  const int lanem = lane & 15;                 // N (and M for A frag)
  const bool hi   = (lane >= 16);              // K half: lo -> K0/K1, hi -> K2/K3
  const int ka    = hi ? 2 : 0;
  const int col0  = (blockIdx.x * 8 + wave) * 16;
  const float mrow = (lanem == 0) ? 1.0f : 0.0f;  // A rows 1..15 are zero

  v8f c = {};
#pragma unroll 8
  for (int kk = 0; kk < DMOD; kk += 4) {
    const v4f sv = *(const v4f*)(s + kk);      // uniform -> s_load_b128
    v2f a, b;
    a.x = (hi ? sv.z : sv.x) * mrow;           // cndmask + mul, no predication
    a.y = (hi ? sv.w : sv.y) * mrow;
    b.x = D2[(size_t)(kk + ka)     * IVOC + col0 + lanem];
    b.y = D2[(size_t)(kk + ka + 1) * IVOC + col0 + lanem];
    // 8 args: (neg_a, A, neg_b, B, c_mod, C, reuse_a, reuse_b)
    c = __builtin_amdgcn_wmma_f32_16x16x4_f32(false, a, false, b,
                                              (short)0, c, false, false);
  }
  if (lane < 16) out[col0 + lane] = c[0];      // C VGPR0, lanes 0-15 = row M=0
}

// ---------------------------------------------------------------------------
extern "C" void kernel_launch(void* const* d_in, const int* in_sizes, int n_in,
                              void* d_out, int out_size, void* d_ws, size_t ws_size,
                              hipStream_t stream) {
  (void)in_sizes; (void)n_in; (void)out_size; (void)ws_size;
  const float* x   = (const float*)d_in[0];
  const float* DE  = (const float*)d_in[1];
  const float* PE  = (const float*)d_in[2];
  const float* D1  = (const float*)d_in[3];
  const float* D2  = (const float*)d_in[4];
  const float* Att = (const float*)d_in[5];
  float* out = (float*)d_out;

  float* partial = (float*)d_ws;               // NPART * 512 floats
  float* sbuf    = partial + NPART * 2 * DMOD; // 256 floats (16B aligned)

  phaseA<<<NPART, 256, 0, stream>>>(x, DE, partial);
  phaseB<<<1, 256, 0, stream>>>(partial, PE, D1, Att, sbuf);
  phaseC<<<(IVOC / 16) / 8, 256, 0, stream>>>(sbuf, D2, out);
}